// Encoder_54150947668719
// MI455X (gfx1250) — compile-verified
//
#include <hip/hip_runtime.h>
#include <hip/hip_bf16.h>

typedef float v2f __attribute__((ext_vector_type(2)));
typedef float v8f __attribute__((ext_vector_type(8)));

#define SLOPE 0.2f

// ---------- helpers ----------
__device__ __forceinline__ unsigned fenc(float f) {
    unsigned b = __float_as_uint(f);
    return (b & 0x80000000u) ? ~b : (b | 0x80000000u);
}
__device__ __forceinline__ float fdec(unsigned e) {
    return __uint_as_float((e & 0x80000000u) ? (e & 0x7fffffffu) : ~e);
}

// ---------- tiny utility kernels ----------
__global__ void k_fill(float* p, float v, int n) {
    int i = blockIdx.x * blockDim.x + threadIdx.x;
    if (i < n) p[i] = v;
}

__global__ void k_init_bias(float* acc, const float* __restrict__ b, int total) {
    int i = blockIdx.x * blockDim.x + threadIdx.x;
    if (i < total) acc[i] = b[i & 127];
}

__global__ void k_relu(float* p, int n) {
    int i = blockIdx.x * blockDim.x + threadIdx.x;
    if (i < n) p[i] = fmaxf(p[i], 0.f);
}

// ---------- self-loop attr (mean of incoming edge_attr) ----------
__global__ void k_loop_acc(const int* __restrict__ ei, const float* __restrict__ eattr,
                           float* cnt, float* lattr, int E_) {
    int e = blockIdx.x * blockDim.x + threadIdx.x;
    if (e >= E_) return;
    int d = ei[E_ + e];
    unsafeAtomicAdd(cnt + d, 1.0f);
    const float* ea = eattr + (long)e * 4;
    float* la = lattr + (long)d * 4;
    unsafeAtomicAdd(la + 0, ea[0]);
    unsafeAtomicAdd(la + 1, ea[1]);
    unsafeAtomicAdd(la + 2, ea[2]);
    unsafeAtomicAdd(la + 3, ea[3]);
}

__global__ void k_loop_div(float* lattr, const float* __restrict__ cnt, int n4) {
    int i = blockIdx.x * blockDim.x + threadIdx.x;
    if (i < n4) lattr[i] = lattr[i] / fmaxf(cnt[i >> 2], 1.0f);
}

// ---------- GEMM via fp32 WMMA: out = X @ W ----------
// One wave computes a 16-row x 128-col strip for ONE weight matrix.
// gridDim.y selects (Wl -> xl) vs (Wr -> xr); 8 v8f accumulators = 64 VGPRs, no spill.
__global__ void k_gemm(const float* __restrict__ X,
                       const float* __restrict__ Wl, const float* __restrict__ Wr,
                       float* __restrict__ xl, float* __restrict__ xr, int nrows) {
    const float* W = (blockIdx.y == 0) ? Wl : Wr;
    float* out     = (blockIdx.y == 0) ? xl : xr;

    int lane = threadIdx.x & 31;
    int wave = threadIdx.x >> 5;
    int rowTile = blockIdx.x * 4 + wave;      // 16-row tile index
    long row0 = (long)rowTile * 16;
    if (row0 >= nrows) return;

    int m = lane & 15;
    int khalf = lane >> 4;                    // 0 or 1
    long row = row0 + m;
    long rowc = row < nrows ? row : (long)nrows - 1;  // clamp (EXEC stays full for WMMA)
    const float* xrow = X + rowc * 128;

    v8f acc[8] = {};
    for (int k0 = 0; k0 < 128; k0 += 4) {
        int k = k0 + khalf * 2;               // A frag: lanes 0-15 hold K=k0..k0+1, 16-31 hold K=k0+2..k0+3
        v2f a;
        a.x = xrow[k];
        a.y = xrow[k + 1];
#pragma unroll
        for (int t = 0; t < 8; ++t) {
            int n = t * 16 + m;
            v2f b;
            b.x = W[k * 128 + n];
            b.y = W[(k + 1) * 128 + n];
            acc[t] = __builtin_amdgcn_wmma_f32_16x16x4_f32(
                false, a, false, b, (short)0, acc[t], false, false);
        }
    }
    // C/D layout: VGPR j -> row = j + khalf*8, col = t*16 + (lane%16)
    float* base = out + row0 * 128 + khalf * (8 * 128) + m;
    if (row0 + 16 <= (long)nrows) {
        // full tile: unguarded stores, immediate offsets from one base address
#pragma unroll
        for (int j = 0; j < 8; ++j) {
#pragma unroll
            for (int t = 0; t < 8; ++t)
                base[j * 128 + t * 16] = acc[t][j];
        }
    } else {
        // tail tile: one guard per row
#pragma unroll
        for (int j = 0; j < 8; ++j) {
            if (row0 + khalf * 8 + j < (long)nrows) {
#pragma unroll
                for (int t = 0; t < 8; ++t)
                    base[j * 128 + t * 16] = acc[t][j];
            }
        }
    }
}

// ---------- per-edge attention logit + segment max ----------
__global__ void k_logit(const float* __restrict__ xl, const float* __restrict__ xr,
                        const int* __restrict__ ei, const float* __restrict__ eattr,
                        const float* __restrict__ lattr, const float* __restrict__ We,
                        const float* __restrict__ att, float* __restrict__ logits,
                        unsigned* menc, int E_, int N_) {
    __shared__ float sWe[4 * 128];
    __shared__ float sAtt[128];
    for (int i = threadIdx.x; i < 512; i += blockDim.x) sWe[i] = We[i];
    for (int i = threadIdx.x; i < 128; i += blockDim.x) sAtt[i] = att[i];
    __syncthreads();

    int lane = threadIdx.x & 31;
    int wid = blockIdx.x * (blockDim.x >> 5) + (threadIdx.x >> 5);
    int ET = E_ + N_;
    if (wid >= ET) return;

    int s, d;
    const float* ea;
    if (wid < E_) { s = ei[wid]; d = ei[E_ + wid]; ea = eattr + (long)wid * 4; }
    else          { s = wid - E_; d = s;           ea = lattr + (long)s * 4; }
    float e0 = ea[0], e1 = ea[1], e2 = ea[2], e3 = ea[3];

    int c = lane * 4;
    float4 vl = *(const float4*)(xl + (long)s * 128 + c);
    float4 vr = *(const float4*)(xr + (long)d * 128 + c);
    const float* pl = (const float*)&vl;
    const float* pr = (const float*)&vr;

    float acc = 0.f;
#pragma unroll
    for (int j = 0; j < 4; ++j) {
        float ee = e0 * sWe[c + j] + e1 * sWe[128 + c + j] +
                   e2 * sWe[256 + c + j] + e3 * sWe[384 + c + j];
        float sv = pl[j] + pr[j] + ee;
        sv = sv > 0.f ? sv : SLOPE * sv;   // leaky_relu
        acc += sv * sAtt[c + j];
    }
    for (int off = 16; off; off >>= 1) acc += __shfl_xor(acc, off, 32);
    if (lane == 0) {
        logits[wid] = acc;
        atomicMax(menc + d, fenc(acc));
    }
}

// ---------- exp(logit - max) and segment sum of denominators ----------
__global__ void k_soft(float* logits, const unsigned* __restrict__ menc, float* den,
                       const int* __restrict__ ei, int E_, int N_) {
    int e = blockIdx.x * blockDim.x + threadIdx.x;
    int ET = E_ + N_;
    if (e >= ET) return;
    int d = (e < E_) ? ei[E_ + e] : (e - E_);
    float ex = expf(logits[e] - fdec(menc[d]));
    logits[e] = ex;
    unsafeAtomicAdd(den + d, ex);
}

// ---------- weighted scatter of messages ----------
__global__ void k_scatter(const float* __restrict__ ex, const float* __restrict__ den,
                          const float* __restrict__ xl, const int* __restrict__ ei,
                          float* acc, int E_, int N_) {
    int lane = threadIdx.x & 31;
    int wid = blockIdx.x * (blockDim.x >> 5) + (threadIdx.x >> 5);
    int ET = E_ + N_;
    if (wid >= ET) return;
    int s, d;
    if (wid < E_) { s = ei[wid]; d = ei[E_ + wid]; }
    else          { s = wid - E_; d = s; }
    float alpha = ex[wid] / (den[d] + 1e-16f);
    int c = lane * 4;
    float4 v = *(const float4*)(xl + (long)s * 128 + c);
    float* out = acc + (long)d * 128 + c;
    unsafeAtomicAdd(out + 0, alpha * v.x);
    unsafeAtomicAdd(out + 1, alpha * v.y);
    unsafeAtomicAdd(out + 2, alpha * v.z);
    unsafeAtomicAdd(out + 3, alpha * v.w);
}

// ---------- driver ----------
extern "C" void kernel_launch(void* const* d_in, const int* in_sizes, int n_in,
                              void* d_out, int out_size, void* d_ws, size_t ws_size,
                              hipStream_t stream) {
    const int* ei      = (const int*)d_in[0];
    const float* eattr = (const float*)d_in[1];
    const float* x0    = (const float*)d_in[2];
    const int E_ = in_sizes[0] / 2;
    const int N_ = in_sizes[2] / 128;
    const int ET = E_ + N_;

    char* ws = (char*)d_ws;
    size_t off = 0;
    auto carve = [&](size_t bytes) -> void* {
        void* p = ws + off;
        off += (bytes + 255) & ~(size_t)255;
        return p;
    };
    float*    xl     = (float*)carve((size_t)N_ * 128 * 4);
    float*    xr     = (float*)carve((size_t)N_ * 128 * 4);
    float*    bufA   = (float*)carve((size_t)N_ * 128 * 4);   // ping buffer (h between layers)
    float*    logits = (float*)carve((size_t)ET * 4);
    unsigned* menc   = (unsigned*)carve((size_t)N_ * 4);
    float*    den    = (float*)carve((size_t)N_ * 4);
    float*    cnt    = (float*)carve((size_t)N_ * 4);
    float*    lattr  = (float*)carve((size_t)N_ * 16);

    const int TB = 256;
    dim3 b256(TB);

    // Self-loop attrs: mean of incoming edge_attr (fill_value='mean')
    k_fill<<<dim3((N_ + TB - 1) / TB), b256, 0, stream>>>(cnt, 0.f, N_);
    k_fill<<<dim3((N_ * 4 + TB - 1) / TB), b256, 0, stream>>>(lattr, 0.f, N_ * 4);
    k_loop_acc<<<dim3((E_ + TB - 1) / TB), b256, 0, stream>>>(ei, eattr, cnt, lattr, E_);
    k_loop_div<<<dim3((N_ * 4 + TB - 1) / TB), b256, 0, stream>>>(lattr, cnt, N_ * 4);

    const int rowTiles = (N_ + 15) / 16;
    const dim3 gGemm((rowTiles + 3) / 4, 2);   // y: 0 -> Wl/xl, 1 -> Wr/xr
    const dim3 gEdgeW((ET + 7) / 8);           // 8 waves per 256-thread block, 1 wave/edge
    const dim3 gEdgeT((ET + TB - 1) / TB);     // 1 thread/edge
    const int total = N_ * 128;

    for (int l = 0; l < 3; ++l) {
        const float* X   = (l == 0) ? x0 : bufA;
        const float* Wl  = (const float*)d_in[3 + l * 5];
        const float* Wr  = (const float*)d_in[4 + l * 5];
        const float* We  = (const float*)d_in[5 + l * 5];
        const float* att = (const float*)d_in[6 + l * 5];
        const float* bia = (const float*)d_in[7 + l * 5];
        float* acc = (l == 2) ? (float*)d_out : bufA;

        // dense transforms (fp32 WMMA); X is fully consumed here, so acc may alias X
        k_gemm<<<gGemm, dim3(128), 0, stream>>>(X, Wl, Wr, xl, xr, N_);

        k_fill<<<dim3((N_ + TB - 1) / TB), b256, 0, stream>>>((float*)menc, 0.f, N_); // enc < all reals
        k_fill<<<dim3((N_ + TB - 1) / TB), b256, 0, stream>>>(den, 0.f, N_);
        k_init_bias<<<dim3((total + TB - 1) / TB), b256, 0, stream>>>(acc, bia, total);

        k_logit<<<gEdgeW, b256, 0, stream>>>(xl, xr, ei, eattr, lattr, We, att,
                                             logits, menc, E_, N_);
        k_soft<<<gEdgeT, b256, 0, stream>>>(logits, menc, den, ei, E_, N_);
        k_scatter<<<gEdgeW, b256, 0, stream>>>(logits, den, xl, ei, acc, E_, N_);

        if (l < 2)
            k_relu<<<dim3((total + TB - 1) / TB), b256, 0, stream>>>(bufA, total);
    }
}